// KernelBasedAttention_6571299963366
// MI455X (gfx1250) — compile-verified
//
#include <hip/hip_runtime.h>
#include <hip/hip_bf16.h>

typedef float v2f __attribute__((ext_vector_type(2)));
typedef float v8f __attribute__((ext_vector_type(8)));

#define B_SZ 4
#define N_SZ 16384
#define D_SZ 256
#define NCHUNK 8
#define CHUNK_N (N_SZ / NCHUNK)   // 2048
#define EPSV 1e-8f

// ---------------------------------------------------------------------------
// Phase 1: partial[b][chunk][d][e] = sum_{n in chunk} relu(k[b][n][d]) * v[b][n][e]
// grid = B * 4 slabs * NCHUNK blocks, 256 threads (8 waves).
// Wave w: d-tile = (w>>1) within 64-row slab, e-half = (w&1)*128 (8 col tiles).
// ---------------------------------------------------------------------------
__global__ __launch_bounds__(256) void la_kv_partial(const float* __restrict__ kmat,
                                                     const float* __restrict__ vmat,
                                                     float* __restrict__ partial) {
    const int idx   = blockIdx.x;
    const int chunk = idx & (NCHUNK - 1);
    const int slab  = (idx >> 3) & 3;
    const int b     = idx >> 5;

    const int tid  = threadIdx.x;
    const int w    = tid >> 5;
    const int lane = tid & 31;
    const int m    = lane & 15;
    const int hi   = lane >> 4;           // 0: K=0,1  1: K=2,3

    const int dtile = w >> 1;             // 0..3 within slab
    const int ebase = (w & 1) * 128;      // 8 tiles of 16 cols
    const int dcol  = slab * 64 + dtile * 16 + m;

    const float* kb = kmat + (size_t)b * N_SZ * D_SZ;
    const float* vb = vmat + (size_t)b * N_SZ * D_SZ;

    v8f acc[8];
#pragma unroll
    for (int t = 0; t < 8; ++t) acc[t] = (v8f){0.f,0.f,0.f,0.f,0.f,0.f,0.f,0.f};

    const int nStart = chunk * CHUNK_N;
    const int nEnd   = nStart + CHUNK_N;

    for (int n0 = nStart; n0 < nEnd; n0 += 4) {
        const int nn = n0 + 2 * hi;
        // A fragment: A[m=d, k=n] = relu(K[n][d]); lanes span d -> coalesced
        const float* kp = kb + (size_t)nn * D_SZ + dcol;
        v2f a;
        a.x = fmaxf(kp[0],    0.f);
        a.y = fmaxf(kp[D_SZ], 0.f);
        // B fragments: B[k=n, e] = V[n][e]
        const float* vp = vb + (size_t)nn * D_SZ + ebase + m;
#pragma unroll
        for (int t = 0; t < 8; ++t) {
            v2f bfr;
            bfr.x = vp[t * 16];
            bfr.y = vp[t * 16 + D_SZ];
            acc[t] = __builtin_amdgcn_wmma_f32_16x16x4_f32(
                false, a, false, bfr, (short)0, acc[t], false, false);
        }
    }

    // store partial slab: C layout — VGPR i: lanes 0-15 -> M=i, lanes 16-31 -> M=i+8
    float* pb = partial + (((size_t)b * NCHUNK + chunk) * D_SZ * D_SZ);
#pragma unroll
    for (int t = 0; t < 8; ++t) {
#pragma unroll
        for (int i = 0; i < 8; ++i) {
            const int drow = slab * 64 + dtile * 16 + (hi ? i + 8 : i);
            pb[(size_t)drow * D_SZ + ebase + t * 16 + m] = acc[t][i];
        }
    }
}

// ---------------------------------------------------------------------------
// Phase 1b: kv[b][d][e] = sum_c partial[b][c][d][e]
// grid = B*D*D/256 = 1024 blocks
// ---------------------------------------------------------------------------
__global__ __launch_bounds__(256) void la_kv_reduce(const float* __restrict__ partial,
                                                    float* __restrict__ kv) {
    const int idx = blockIdx.x * 256 + threadIdx.x;     // 0 .. B*D*D-1
    const int b   = idx >> 16;
    const int rem = idx & 0xFFFF;
    float s = 0.f;
#pragma unroll
    for (int c = 0; c < NCHUNK; ++c)
        s += partial[(((size_t)(b * NCHUNK + c)) << 16) + rem];
    kv[((size_t)b << 16) + rem] = s;
}

// ---------------------------------------------------------------------------
// Phase 2: out[b][n][e] = (sum_d relu(q[b][n][d]) * kv[b][d][e]) / (rowsum+eps)
// grid = B * N/32 = 2048 blocks, 256 threads (8 waves).
// WG stages 32 relu(Q) rows in LDS (pitch 260 dwords), computes row recip,
// wave w: row-tile = w>>2, col base = (w&3)*64 (4 tiles of 16).
// ---------------------------------------------------------------------------
#define QPITCH 260

__global__ __launch_bounds__(256) void la_out(const float* __restrict__ qmat,
                                              const float* __restrict__ kv,
                                              float* __restrict__ out) {
    __shared__ float ldsq[32 * QPITCH];
    __shared__ float dpart[256];
    __shared__ float drcp[32];

    const int b    = blockIdx.x >> 9;
    const int n0   = (blockIdx.x & 511) * 32;
    const int tid  = threadIdx.x;

    // ---- stage relu(Q) block + row-sum partials (coalesced b128 loads) ----
    {
        const int row  = tid >> 3;            // 0..31
        const int cseg = (tid & 7) * 32;      // 8 threads per row
        const float* qrow = qmat + ((size_t)(b * N_SZ + n0 + row)) * D_SZ + cseg;
        float s = 0.f;
#pragma unroll
        for (int j = 0; j < 32; j += 4) {
            float4 x = *(const float4*)(qrow + j);
            x.x = fmaxf(x.x, 0.f);
            x.y = fmaxf(x.y, 0.f);
            x.z = fmaxf(x.z, 0.f);
            x.w = fmaxf(x.w, 0.f);
            *(float4*)&ldsq[row * QPITCH + cseg + j] = x;
            s += x.x + x.y + x.z + x.w;
        }
        dpart[tid] = s;
    }
    __syncthreads();
    if (tid < 32) {
        float s = EPSV;
#pragma unroll
        for (int j = 0; j < 8; ++j) s += dpart[tid * 8 + j];
        drcp[tid] = 1.0f / s;
    }
    __syncthreads();

    // ---- WMMA main loop ----
    const int w    = tid >> 5;
    const int lane = tid & 31;
    const int m    = lane & 15;
    const int hi   = lane >> 4;
    const int rt   = w >> 2;              // row tile 0/1
    const int cb   = (w & 3) * 64;        // 4 col tiles of 16

    v8f acc[4];
#pragma unroll
    for (int t = 0; t < 4; ++t) acc[t] = (v8f){0.f,0.f,0.f,0.f,0.f,0.f,0.f,0.f};

    const float* kvb = kv + ((size_t)b << 16);

    for (int k0 = 0; k0 < D_SZ; k0 += 4) {
        const int kk = k0 + 2 * hi;
        // A fragment from LDS (relu'd Q): contiguous pair -> ds_load_b64
        const float* ap = &ldsq[(rt * 16 + m) * QPITCH + kk];
        v2f a;
        a.x = ap[0];
        a.y = ap[1];
#pragma unroll
        for (int t = 0; t < 4; ++t) {
            const float* bp = kvb + (size_t)kk * D_SZ + cb + t * 16 + m;
            v2f bfr;
            bfr.x = bp[0];
            bfr.y = bp[D_SZ];
            acc[t] = __builtin_amdgcn_wmma_f32_16x16x4_f32(
                false, a, false, bfr, (short)0, acc[t], false, false);
        }
    }

    // ---- epilogue: scale by reciprocal denominator, store ----
    float r[8];
#pragma unroll
    for (int i = 0; i < 8; ++i) r[i] = drcp[rt * 16 + (hi ? i + 8 : i)];

#pragma unroll
    for (int t = 0; t < 4; ++t) {
#pragma unroll
        for (int i = 0; i < 8; ++i) {
            const int rowl = rt * 16 + (hi ? i + 8 : i);
            out[((size_t)(b * N_SZ + n0 + rowl)) * D_SZ + cb + t * 16 + m] =
                acc[t][i] * r[i];
        }
    }
}

// ---------------------------------------------------------------------------
extern "C" void kernel_launch(void* const* d_in, const int* in_sizes, int n_in,
                              void* d_out, int out_size, void* d_ws, size_t ws_size,
                              hipStream_t stream) {
    const float* q = (const float*)d_in[0];
    const float* k = (const float*)d_in[1];
    const float* v = (const float*)d_in[2];
    float* out = (float*)d_out;

    float* kvbuf   = (float*)d_ws;                        // B*D*D      = 1 MB
    float* partial = kvbuf + (size_t)B_SZ * D_SZ * D_SZ;  // B*8*D*D    = 8 MB

    la_kv_partial<<<B_SZ * 4 * NCHUNK, 256, 0, stream>>>(k, v, partial);
    la_kv_reduce<<<(B_SZ * D_SZ * D_SZ) / 256, 256, 0, stream>>>(partial, kvbuf);
    la_out<<<B_SZ * (N_SZ / 32), 256, 0, stream>>>(q, kvbuf, out);
}